// REGCN_23278722744746
// MI455X (gfx1250) — compile-verified
//
#include <hip/hip_runtime.h>

typedef __attribute__((ext_vector_type(2))) float v2f;
typedef __attribute__((ext_vector_type(8))) float v8f;

#define D 128  // D_IN == D_OUT == 128

// ---------------------------------------------------------------------------
// 0) zero deg, compute rw = leaky_relu(relation_weight * 100)
// ---------------------------------------------------------------------------
__global__ void regcn_init_kernel(const float* __restrict__ relw,
                                  float* __restrict__ rw,
                                  float* __restrict__ deg,
                                  int n_dst, int n_types) {
    int i = blockIdx.x * blockDim.x + threadIdx.x;
    if (i < n_dst) deg[i] = 0.0f;
    if (i < n_types) {
        float s = relw[i] * 100.0f;
        rw[i] = (s > 0.0f) ? s : 0.01f * s;   // leaky_relu, default slope
    }
}

// ---------------------------------------------------------------------------
// 1) deg[col[e]] += rw[etype[e]]   (native f32 atomics; deg lives in L2)
// ---------------------------------------------------------------------------
__global__ void regcn_deg_kernel(const int* __restrict__ col,
                                 const int* __restrict__ etype,
                                 const float* __restrict__ rw,
                                 float* __restrict__ deg, int E) {
    int e = blockIdx.x * blockDim.x + threadIdx.x;
    if (e < E) unsafeAtomicAdd(&deg[col[e]], rw[etype[e]]);
}

// ---------------------------------------------------------------------------
// 2) H = x_src @ W  via V_WMMA_F32_16X16X4_F32 (fp32 stays exact vs reference)
//    block = 256 threads = 8 waves; block covers 16 rows, wave w covers
//    columns [16w, 16w+16). K loop fully unrolled -> 32 wmma per wave.
// ---------------------------------------------------------------------------
__global__ void __launch_bounds__(256)
regcn_gemm_kernel(const float* __restrict__ A,   // [M, 128] x_src
                  const float* __restrict__ B,   // [128, 128] weight
                  float* __restrict__ H,         // [M, 128] h_src
                  int M) {
    const int wave = threadIdx.x >> 5;
    const int lane = threadIdx.x & 31;
    const int half = lane >> 4;        // 0: lanes 0-15, 1: lanes 16-31
    const int l15  = lane & 15;
    const int m0   = blockIdx.x * 16;
    const int n0   = wave * 16;

    // clamp row for A loads so EXEC stays all-1s even on a ragged tail
    const int arow = (m0 + l15 < M) ? (m0 + l15) : (M - 1);
    const float* Ap = A + (size_t)arow * D;

    v8f c = {};
#pragma unroll
    for (int k0 = 0; k0 < D; k0 += 4) {
        const int kk = k0 + 2 * half;           // ISA 16x4 f32 A layout
        v2f a = *(const v2f*)(Ap + kk);         // A[m, kk], A[m, kk+1]
        v2f b;
        b.x = B[(size_t)kk       * D + n0 + l15];  // B[kk,   n]
        b.y = B[(size_t)(kk + 1) * D + n0 + l15];  // B[kk+1, n]
        c = __builtin_amdgcn_wmma_f32_16x16x4_f32(
                /*neg_a=*/false, a, /*neg_b=*/false, b,
                /*c_mod=*/(short)0, c, /*reuse_a=*/false, /*reuse_b=*/false);
    }

    // C/D layout: VGPR r -> row m0 + r + 8*half, lanes l15 -> col n0+l15
    const int rbase = m0 + 8 * half;
#pragma unroll
    for (int r = 0; r < 8; ++r) {
        int m = rbase + r;
        if (m < M) H[(size_t)m * D + n0 + l15] = c[r];
    }
}

// ---------------------------------------------------------------------------
// 3) out[i,:] = bias  (one thread per 4 features)
// ---------------------------------------------------------------------------
__global__ void regcn_bias_kernel(const float* __restrict__ bias,
                                  float* __restrict__ out, int n32) {
    int i = blockIdx.x * blockDim.x + threadIdx.x;   // over n_dst * 32
    if (i < n32) {
        int d4 = (i & 31) * 4;
        float4 b = *(const float4*)(bias + d4);
        *(float4*)(out + (size_t)(i >> 5) * D + d4) = b;
    }
}

// ---------------------------------------------------------------------------
// 4) out[col[e],:] += (rw[etype]/deg[col]) * H[row[e],:]
//    one thread per (edge, 4 features): float4 gather + 4 native f32 atomics
// ---------------------------------------------------------------------------
__global__ void regcn_scatter_kernel(const int* __restrict__ row,
                                     const int* __restrict__ col,
                                     const int* __restrict__ etype,
                                     const float* __restrict__ rw,
                                     const float* __restrict__ deg,
                                     const float* __restrict__ H,
                                     float* __restrict__ out, int E) {
    int i = blockIdx.x * blockDim.x + threadIdx.x;   // over E * 32
    if (i >= E * 32) return;
    int e  = i >> 5;
    int d4 = (i & 31) * 4;
    int r  = row[e];
    int cd = col[e];
    float ew = rw[etype[e]] / deg[cd];
    float4 h = *(const float4*)(H + (size_t)r * D + d4);
    float* o = out + (size_t)cd * D + d4;
    unsafeAtomicAdd(o + 0, ew * h.x);
    unsafeAtomicAdd(o + 1, ew * h.y);
    unsafeAtomicAdd(o + 2, ew * h.z);
    unsafeAtomicAdd(o + 3, ew * h.w);
}

// ---------------------------------------------------------------------------
extern "C" void kernel_launch(void* const* d_in, const int* in_sizes, int n_in,
                              void* d_out, int out_size, void* d_ws, size_t ws_size,
                              hipStream_t stream) {
    const float* x_src  = (const float*)d_in[0];
    // d_in[1] (x_target) is unused by the reference's output path
    const int*   row    = (const int*)d_in[2];
    const int*   col    = (const int*)d_in[3];
    const int*   etype  = (const int*)d_in[4];
    const float* weight = (const float*)d_in[5];
    const float* bias   = (const float*)d_in[6];
    const float* relw   = (const float*)d_in[7];

    const int n_src   = in_sizes[0] / D;   // 200000
    const int n_dst   = in_sizes[1] / D;   // 100000
    const int E       = in_sizes[2];       // 600000
    const int n_types = in_sizes[7];       // 7

    // workspace: rw (256B) | deg (n_dst f32, 256B aligned) | H (n_src*128 f32)
    char*  ws    = (char*)d_ws;
    float* rw    = (float*)ws;
    float* deg   = (float*)(ws + 256);
    size_t deg_b = (((size_t)n_dst * sizeof(float)) + 255) & ~(size_t)255;
    float* H     = (float*)(ws + 256 + deg_b);

    float* out = (float*)d_out;

    regcn_init_kernel<<<(n_dst + 255) / 256, 256, 0, stream>>>(relw, rw, deg,
                                                               n_dst, n_types);
    regcn_deg_kernel<<<(E + 255) / 256, 256, 0, stream>>>(col, etype, rw, deg, E);

    int mtiles = (n_src + 15) / 16;        // 12500
    regcn_gemm_kernel<<<mtiles, 256, 0, stream>>>(x_src, weight, H, n_src);

    long long nb = (long long)n_dst * 32;
    regcn_bias_kernel<<<(int)((nb + 255) / 256), 256, 0, stream>>>(bias, out, (int)nb);

    long long ns = (long long)E * 32;
    regcn_scatter_kernel<<<(int)((ns + 255) / 256), 256, 0, stream>>>(
        row, col, etype, rw, deg, H, out, E);
}